// DGCNAgg_27152783245805
// MI455X (gfx1250) — compile-verified
//
#include <hip/hip_runtime.h>

typedef float v2f __attribute__((ext_vector_type(2)));
typedef float v8f __attribute__((ext_vector_type(8)));

// ---------------------------------------------------------------- utilities
__global__ void fill_f32_kernel(float* __restrict__ p, float v, int n) {
  int i = blockIdx.x * blockDim.x + threadIdx.x;
  int st = gridDim.x * blockDim.x;
  for (; i < n; i += st) p[i] = v;
}

// ------------------------------------------------------- column mean / std
__global__ void col_stats_kernel(const float* __restrict__ x,
                                 float* __restrict__ gsum,
                                 float* __restrict__ gsq,
                                 int N, int F) {
  __shared__ float ssum[128];
  __shared__ float ssq[128];
  for (int c = threadIdx.x; c < F; c += blockDim.x) { ssum[c] = 0.f; ssq[c] = 0.f; }
  __syncthreads();
  long long total = (long long)N * F;
  long long i  = (long long)blockIdx.x * blockDim.x + threadIdx.x;
  long long st = (long long)gridDim.x * blockDim.x;
  for (; i < total; i += st) {
    float v = x[i];
    int c = (int)(i & (long long)(F - 1));  // F is a power of two (128)
    atomicAdd(&ssum[c], v);
    atomicAdd(&ssq[c], v * v);
  }
  __syncthreads();
  for (int c = threadIdx.x; c < F; c += blockDim.x) {
    atomicAdd(&gsum[c], ssum[c]);
    atomicAdd(&gsq[c],  ssq[c]);
  }
}

__global__ void col_finalize_kernel(const float* __restrict__ gsum,
                                    const float* __restrict__ gsq,
                                    float* __restrict__ mean,
                                    float* __restrict__ rstd,
                                    int N, int F) {
  int c = blockIdx.x * blockDim.x + threadIdx.x;
  if (c < F) {
    float s = gsum[c];
    float m = s / (float)N;
    float var = (gsq[c] - s * m) / (float)(N - 1);   // ddof = 1
    mean[c] = m;
    rstd[c] = rsqrtf(var);
  }
}

__global__ void normalize_kernel(const float* __restrict__ x,
                                 const float* __restrict__ mean,
                                 const float* __restrict__ rstd,
                                 float* __restrict__ xn,
                                 long long total, int F) {
  long long i  = (long long)blockIdx.x * blockDim.x + threadIdx.x;
  long long st = (long long)gridDim.x * blockDim.x;
  for (; i < total; i += st) {
    int c = (int)(i & (long long)(F - 1));
    xn[i] = (x[i] - mean[c]) * rstd[c];
  }
}

// ------------------------------------------------------------ degrees/norm
__global__ void deg_accum_kernel(const float* __restrict__ ea,
                                 const int* __restrict__ dst,
                                 float* __restrict__ deg,
                                 int E, int N) {
  int e  = blockIdx.x * blockDim.x + threadIdx.x;
  int st = gridDim.x * blockDim.x;
  for (; e < E; e += st) {
    int d = dst[e];
    atomicAdd(&deg[d],         fabsf(ea[e * 3 + 0]));
    atomicAdd(&deg[N + d],     fabsf(ea[e * 3 + 1]));
    atomicAdd(&deg[2 * N + d], fabsf(ea[e * 3 + 2]));
  }
}

__global__ void deg_finalize_kernel(const float* __restrict__ deg,
                                    float* __restrict__ dis,
                                    float* __restrict__ invd,
                                    int n) {
  int i  = blockIdx.x * blockDim.x + threadIdx.x;
  int st = gridDim.x * blockDim.x;
  for (; i < n; i += st) {
    float dv = deg[i];
    dis[i]  = rsqrtf(dv);
    invd[i] = 1.0f / dv;
  }
}

__global__ void edge_norm_kernel(const float* __restrict__ ea,
                                 const int* __restrict__ src,
                                 const int* __restrict__ dst,
                                 const float* __restrict__ dis,
                                 float* __restrict__ norm,
                                 int E, int N) {
  int e  = blockIdx.x * blockDim.x + threadIdx.x;
  int st = gridDim.x * blockDim.x;
  for (; e < E; e += st) {
    int s = src[e], d = dst[e];
#pragma unroll
    for (int t = 0; t < 3; ++t) {
      norm[(size_t)t * E + e] = dis[t * N + s] * fabsf(ea[e * 3 + t]) * dis[t * N + d];
    }
  }
}

// ------------------------------------------------------------- fp32 WMMA GEMM
// C[M,Nc] = A[M,K] @ B[K,Nc]; each wave owns a 16x64 strip of C.
// V_WMMA_F32_16X16X4_F32: A=16x4 (2 VGPR), B=4x16 (2 VGPR), C/D=16x16 (8 VGPR).
__global__ void gemm_wmma_kernel(const float* __restrict__ A,
                                 const float* __restrict__ B,
                                 float* __restrict__ C,
                                 int M, int K, int Nc) {
  const int lane   = threadIdx.x & 31;
  const int wave   = (blockIdx.x * blockDim.x + threadIdx.x) >> 5;
  const int tilesN = Nc >> 6;                 // 64-wide strips
  const int tm = wave / tilesN;
  const int tn = wave % tilesN;
  if (tm * 16 >= M) return;                   // wave-uniform

  const int mrow = tm * 16 + (lane & 15);
  const int ncol = tn * 64 + (lane & 15);
  const int kb   = (lane >> 4) << 1;          // 0 or 2: K-pair per half-wave
  const float* __restrict__ a_ptr = A + (size_t)mrow * K + kb;
  const float* __restrict__ b_ptr = B + (size_t)kb * Nc + ncol;

  v8f c0 = {}, c1 = {}, c2 = {}, c3 = {};
  for (int k = 0; k < K; k += 4) {
    v2f a;
    a[0] = a_ptr[k];
    a[1] = a_ptr[k + 1];
    const float* bp = b_ptr + (size_t)k * Nc;
    v2f b0, b1, b2, b3;
    b0[0] = bp[0];       b0[1] = bp[Nc];
    b1[0] = bp[16];      b1[1] = bp[Nc + 16];
    b2[0] = bp[32];      b2[1] = bp[Nc + 32];
    b3[0] = bp[48];      b3[1] = bp[Nc + 48];
    c0 = __builtin_amdgcn_wmma_f32_16x16x4_f32(false, a, false, b0, (short)0, c0, false, false);
    c1 = __builtin_amdgcn_wmma_f32_16x16x4_f32(false, a, false, b1, (short)0, c1, false, false);
    c2 = __builtin_amdgcn_wmma_f32_16x16x4_f32(false, a, false, b2, (short)0, c2, false, false);
    c3 = __builtin_amdgcn_wmma_f32_16x16x4_f32(false, a, false, b3, (short)0, c3, false, false);
  }

  const int row0 = tm * 16 + ((lane >> 4) << 3);
  float* crow = C + (size_t)row0 * Nc + tn * 64 + (lane & 15);
#pragma unroll
  for (int v = 0; v < 8; ++v) {
    crow[(size_t)v * Nc]      = c0[v];
    crow[(size_t)v * Nc + 16] = c1[v];
    crow[(size_t)v * Nc + 32] = c2[v];
    crow[(size_t)v * Nc + 48] = c3[v];
  }
}

// ------------------------------------------------------- aggregation kernels
__global__ void self_init_kernel(const float* __restrict__ h,
                                 const float* __restrict__ invd,
                                 float* __restrict__ agg,
                                 long long total, int Dlog) {
  long long i  = (long long)blockIdx.x * blockDim.x + threadIdx.x;
  long long st = (long long)gridDim.x * blockDim.x;
  for (; i < total; i += st) {
    agg[i] = invd[i >> Dlog] * h[i];
  }
}

__global__ void scatter_kernel(const float* __restrict__ h,
                               const float* __restrict__ norm,
                               const int* __restrict__ src,
                               const int* __restrict__ dst,
                               float* __restrict__ agg,
                               int E, int D, int chunkLog) {
  const int chunks = D >> 2;                       // float4 per thread
  long long total  = (long long)E * chunks;
  long long i  = (long long)blockIdx.x * blockDim.x + threadIdx.x;
  long long st = (long long)gridDim.x * blockDim.x;
  for (; i < total; i += st) {
    int e = (int)(i >> chunkLog);
    int j = ((int)i & (chunks - 1)) << 2;
    float w = norm[e];
    int s = src[e], d = dst[e];
    const float4 hv = *(const float4*)(h + (size_t)s * D + j);
    float* ap = agg + (size_t)d * D + j;
    atomicAdd(ap + 0, w * hv.x);
    atomicAdd(ap + 1, w * hv.y);
    atomicAdd(ap + 2, w * hv.z);
    atomicAdd(ap + 3, w * hv.w);
  }
}

__global__ void bias_relu_kernel(const float* __restrict__ agg,
                                 const float* __restrict__ bias,
                                 float* __restrict__ out,
                                 long long total, int D) {
  long long i  = (long long)blockIdx.x * blockDim.x + threadIdx.x;
  long long st = (long long)gridDim.x * blockDim.x;
  for (; i < total; i += st) {
    int d = (int)(i & (long long)(D - 1));
    out[i] = fmaxf(agg[i] + bias[d], 0.f);
  }
}

// fused bias + relu + (B,S,NN,D)->(B,NN,S,D) permute + concat-over-types store
__global__ void final_out_kernel(const float* __restrict__ agg,
                                 const float* __restrict__ bias,
                                 float* __restrict__ out,
                                 int N, int D, int Dlog, int S, int NN,
                                 int TD, int toff) {
  long long total = (long long)N * D;
  long long i  = (long long)blockIdx.x * blockDim.x + threadIdx.x;
  long long st = (long long)gridDim.x * blockDim.x;
  for (; i < total; i += st) {
    int n = (int)(i >> Dlog);
    int d = (int)(i & (long long)(D - 1));
    int b   = n / (S * NN);
    int rem = n - b * (S * NN);
    int s   = rem / NN;
    int nn  = rem - s * NN;
    size_t o = (((size_t)(b * NN + nn) * S + s) * TD) + toff + d;
    out[o] = fmaxf(agg[i] + bias[d], 0.f);
  }
}

// ---------------------------------------------------------------- launcher
extern "C" void kernel_launch(void* const* d_in, const int* in_sizes, int n_in,
                              void* d_out, int out_size, void* d_ws, size_t ws_size,
                              hipStream_t stream) {
  const float* x  = (const float*)d_in[0];
  const float* ea = (const float*)d_in[1];
  const float* W1 = (const float*)d_in[2];
  const float* b1 = (const float*)d_in[3];
  const float* W2 = (const float*)d_in[4];
  const float* b2 = (const float*)d_in[5];
  const int*   ei = (const int*)d_in[6];

  const int T  = 3;                       // edge types
  const int F  = 128;                     // input features
  const int N  = in_sizes[0] / F;         // 32768 nodes
  const int E  = in_sizes[1] / T;         // 524288 edges
  const int D1 = in_sizes[2] / (T * F);   // 256
  const int D2 = in_sizes[4] / (T * D1);  // 128
  const int S = 16, NN = 512;

  const int* srcIdx = ei;
  const int* dstIdx = ei + E;

  // workspace layout (floats)
  float* ws   = (float*)d_ws;
  float* xn   = ws;                              // N*F
  float* mean = xn   + (size_t)N * F;            // F
  float* rstd = mean + F;                        // F
  float* gsum = rstd + F;                        // F
  float* gsq  = gsum + F;                        // F
  float* deg  = gsq  + F;                        // 3N
  float* dis  = deg  + (size_t)3 * N;            // 3N
  float* invd = dis  + (size_t)3 * N;            // 3N
  float* norm = invd + (size_t)3 * N;            // 3E
  float* h    = norm + (size_t)3 * E;            // N*D1
  float* agg  = h    + (size_t)N * D1;           // N*D1 (reused: h2 | agg2)

  float* out = (float*)d_out;

  const int BT = 256;

  // ---- normalization of x ----
  fill_f32_kernel<<<1, BT, 0, stream>>>(gsum, 0.f, 2 * F);
  col_stats_kernel<<<512, BT, 0, stream>>>(x, gsum, gsq, N, F);
  col_finalize_kernel<<<1, 128, 0, stream>>>(gsum, gsq, mean, rstd, N, F);
  normalize_kernel<<<4096, BT, 0, stream>>>(x, mean, rstd, xn, (long long)N * F, F);

  // ---- degrees + per-edge symmetric norms for all 3 types ----
  fill_f32_kernel<<<128, BT, 0, stream>>>(deg, 1.0f, 3 * N);   // self-loop weight 1
  deg_accum_kernel<<<2048, BT, 0, stream>>>(ea, dstIdx, deg, E, N);
  deg_finalize_kernel<<<128, BT, 0, stream>>>(deg, dis, invd, 3 * N);
  edge_norm_kernel<<<2048, BT, 0, stream>>>(ea, srcIdx, dstIdx, dis, norm, E, N);

  const int D1log = __builtin_ctz(D1);
  const int D2log = __builtin_ctz(D2);
  const int cl1   = __builtin_ctz(D1 >> 2);
  const int cl2   = __builtin_ctz(D2 >> 2);

  for (int t = 0; t < T; ++t) {
    const float* W1t = W1 + (size_t)t * F * D1;
    const float* b1t = b1 + (size_t)t * D1;
    const float* W2t = W2 + (size_t)t * D1 * D2;
    const float* b2t = b2 + (size_t)t * D2;
    const float* nmt = norm + (size_t)t * E;
    const float* idt = invd + (size_t)t * N;

    // conv1: h = xn @ W1[t]; agg = invd*h + scatter(norm*h[src]); h = relu(agg + b1)
    {
      int waves = (N / 16) * (D1 / 64);
      gemm_wmma_kernel<<<waves / 8, BT, 0, stream>>>(xn, W1t, h, N, F, D1);
      self_init_kernel<<<8192, BT, 0, stream>>>(h, idt, agg, (long long)N * D1, D1log);
      scatter_kernel<<<16384, BT, 0, stream>>>(h, nmt, srcIdx, dstIdx, agg, E, D1, cl1);
      bias_relu_kernel<<<8192, BT, 0, stream>>>(agg, b1t, h, (long long)N * D1, D1);
    }

    // conv2: h2 = h @ W2[t]; agg2 = invd*h2 + scatter; out = relu(agg2 + b2) permuted
    float* h2   = agg;                       // agg buffer now dead; reuse halves
    float* agg2 = agg + (size_t)N * D2;
    {
      int waves = (N / 16) * (D2 / 64);
      gemm_wmma_kernel<<<waves / 8, BT, 0, stream>>>(h, W2t, h2, N, D1, D2);
      self_init_kernel<<<4096, BT, 0, stream>>>(h2, idt, agg2, (long long)N * D2, D2log);
      scatter_kernel<<<8192, BT, 0, stream>>>(h2, nmt, srcIdx, dstIdx, agg2, E, D2, cl2);
      final_out_kernel<<<4096, BT, 0, stream>>>(agg2, b2t, out, N, D2, D2log, S, NN,
                                                T * D2, t * D2);
    }
  }
}